// Mann_ELT_16750372455095
// MI455X (gfx1250) — compile-verified
//
#include <hip/hip_runtime.h>
#include <math.h>

// ---------------------------------------------------------------------------
// Mann eddy-lifetime tau(k) kernel for gfx1250 (MI455X).
// Elementwise transcendental map: memory-bound target ~11.5us @ 23.3 TB/s.
// No matmul structure -> no WMMA. CDNA5 paths used instead:
//   * GLOBAL_LOAD_ASYNC_TO_LDS_B128 double-buffered prefetch (ASYNCcnt)
//   * s_wait_asynccnt split-counter sync
//   * ds_load_b128 tile reads, global_store_b128 streaming stores
// ---------------------------------------------------------------------------

#define DEV __device__ __forceinline__

#if defined(__HIP_DEVICE_COMPILE__)
#define MANN_DEVICE_PASS 1
#else
#define MANN_DEVICE_PASS 0
#endif

namespace mann {

constexpr float  LS2         = 0.59f * 0.59f;          // LENGTH_SCALE^2
constexpr float  TIME_SCALE  = 3.9f;
constexpr float  INVPHI      = 0.61803398874989485f;   // 1/phi (branch split in s)
constexpr float  LOG2_INVPHI = -0.69424191363113475f;  // log2(1/phi)
// A = G(4/3)G(5/2)/(G(17/6)G(1)),  B = G(4/3)G(-5/2)/(G(1/3)G(-3/2)) = -2/15 exactly
constexpr float  A_COEF = 0.6883439f;
constexpr float  B_COEF = -2.0f / 15.0f;

// Series truncation: args bounded by 0.618 -> geometric convergence.
constexpr int NA = 24;   // branch-A terms ~ n^-3.5 * 0.618^n  -> ~1e-10 tail
constexpr int NB = 40;   // branch-B terms ~ n^0.83 * 0.618^n  -> ~1e-7 tail

struct PolyA_t { float c[NA + 1]; };
struct PolyB_t { float c[NB + 1]; };

// Compile-time Pochhammer-ratio coefficients: C_{n+1} = C_n*(a+n)(b+n)/((c+n)(n+1))
constexpr PolyA_t makeA() {
  PolyA_t p{};
  double a = 1.0 / 3.0, b = -3.0 / 2.0, c = 4.0 / 3.0, term = 1.0;
  p.c[0] = 1.0f;
  for (int n = 0; n < NA; ++n) {
    term *= (a + n) * (b + n) / ((c + n) * (n + 1.0));
    p.c[n + 1] = (float)term;
  }
  return p;
}
constexpr PolyB_t makeB() {
  PolyB_t p{};
  double a = 17.0 / 6.0, b = 5.0 / 2.0, c = 7.0 / 2.0, term = 1.0;
  p.c[0] = 1.0f;
  for (int n = 0; n < NB; ++n) {
    term *= (a + n) * (b + n) / ((c + n) * (n + 1.0));
    p.c[n + 1] = (float)term;
  }
  return p;
}
constexpr PolyA_t kA = makeA();
constexpr PolyB_t kB = makeB();

// Fast hardware transcendentals (v_log_f32 / v_exp_f32 / v_rcp_f32 / v_rsq_f32)
DEV float fexp2(float x) {
#if MANN_DEVICE_PASS && __has_builtin(__builtin_amdgcn_exp2f)
  return __builtin_amdgcn_exp2f(x);
#else
  return exp2f(x);
#endif
}
DEV float flog2(float x) {
#if MANN_DEVICE_PASS && __has_builtin(__builtin_amdgcn_logf)
  return __builtin_amdgcn_logf(x);
#else
  return log2f(x);
#endif
}
DEV float frcp(float x) {
#if MANN_DEVICE_PASS && __has_builtin(__builtin_amdgcn_rcpf)
  return __builtin_amdgcn_rcpf(x);
#else
  return 1.0f / x;
#endif
}
DEV float frsq(float x) {
#if MANN_DEVICE_PASS && __has_builtin(__builtin_amdgcn_rsqf)
  return __builtin_amdgcn_rsqf(x);
#else
  return 1.0f / sqrtf(x);
#endif
}

// tau for one element given s = (LENGTH_SCALE*|k|)^2, t = 1/s.
// Branch A (s>=1/phi): Pfaff: f = (1-w)^{1/3}*PolyA(w), w=1/(1+max(s,1/phi)).
// Branch B (s< 1/phi): f = A*sB^{1/3} + B*sB^{17/6}*PolyB(-sB), sB=min(s,1/phi),
//   log2(sB)=min(log2 s, log2(1/phi)) (monotonicity: no second log).
// Result: 3.9 * s^{-1/3} / sqrt(f).  Branchless select (lanes ~60/40 mixed).
DEV float tau_of_s(float s) {
  const float ls = flog2(s);

  const float sA = fmaxf(s, INVPHI);
  const float w  = frcp(1.0f + sA);                    // in [0, 0.618]
  const float pa = fexp2(flog2(1.0f - w) * (1.0f / 3.0f));
  float accA = kA.c[NA];
#pragma unroll
  for (int i = NA - 1; i >= 0; --i) accA = __builtin_fmaf(accA, w, kA.c[i]);
  const float fa = pa * accA;

  const float lsB = fminf(ls, LOG2_INVPHI);
  const float u   = -fminf(s, INVPHI);                 // in (-0.618, 0]
  float accB = kB.c[NB];
#pragma unroll
  for (int i = NB - 1; i >= 0; --i) accB = __builtin_fmaf(accB, u, kB.c[i]);
  const float fb = __builtin_fmaf(B_COEF * fexp2(lsB * (17.0f / 6.0f)), accB,
                                  A_COEF * fexp2(lsB * (1.0f / 3.0f)));

  const float f = (s >= INVPHI) ? fa : fb;
  return TIME_SCALE * fexp2(ls * (-1.0f / 3.0f)) * frsq(f);
}

// Generic pointer -> LDS byte offset (as(3) pointers are 32-bit; their value
// IS the wave-relative LDS offset used by async-to-LDS VDST).
typedef __attribute__((address_space(3))) const void lds_cvoid;
DEV unsigned lds_off_of(const void* p) {
  return (unsigned)(unsigned long long)(lds_cvoid*)p;
}

// Async DMA: 16B global -> LDS, tracked by ASYNCcnt (no VGPRs held in flight).
DEV void async_load_b128(unsigned lds_off, const float* sbase, unsigned voff) {
#if MANN_DEVICE_PASS
  asm volatile("global_load_async_to_lds_b128 %0, %1, %2 offset:0"
               :: "v"(lds_off), "v"(voff), "s"(sbase)
               : "memory");
#else
  (void)lds_off; (void)sbase; (void)voff;
#endif
}

template <int N> DEV void wait_asynccnt() {
#if MANN_DEVICE_PASS && __has_builtin(__builtin_amdgcn_s_wait_asynccnt)
  __builtin_amdgcn_s_wait_asynccnt(N);
#elif MANN_DEVICE_PASS
  asm volatile("s_wait_asynccnt %0" :: "i"(N) : "memory");
#endif
}

} // namespace mann

constexpr int TPB = 256;

// Each thread owns one 48B pack (4 elements) per pipeline stage.
// Stage i+1 is async-DMA'd into the alternate LDS buffer while stage i's
// ~64-FMA chains execute; s_wait_asynccnt 3 retires the previous stage.
// SoA tile layout -> 16B lane stride on ds_load_b128 (bank-conflict-free).
__global__ __launch_bounds__(TPB) void mann_tau_kernel(
    const float* __restrict__ k, float* __restrict__ out,
    long long npack, long long n) {
  using namespace mann;
  __shared__ float4 tile[2][3][TPB];

  const int tid = threadIdx.x;
  const long long stride = (long long)gridDim.x * blockDim.x;
  const long long g      = (long long)blockIdx.x * blockDim.x + tid;
  float4* __restrict__ ov = (float4*)out;

  if (npack > 0) {
    const long long maxp = npack - 1;

    // Prologue: stage this thread's first pack (clamped: loads are pure reads,
    // clamping keeps per-wave ASYNCcnt issue counts uniform).
    {
      const long long p0 = g < maxp ? g : maxp;
      const unsigned base = (unsigned)(p0 * 48ll);
#pragma unroll
      for (int j = 0; j < 3; ++j)
        async_load_b128(lds_off_of(&tile[0][j][tid]), k, base + 16u * j);
    }

    int phase = 0;
    for (long long p = g; p < npack; p += stride) {
      // Issue next stage into the alternate buffer (clamped at the end).
      long long pn = p + stride;
      if (pn > maxp) pn = maxp;
      const unsigned nbase = (unsigned)(pn * 48ll);
#pragma unroll
      for (int j = 0; j < 3; ++j)
        async_load_b128(lds_off_of(&tile[phase ^ 1][j][tid]), k, nbase + 16u * j);

      // 3 newly issued remain outstanding; the 3 for this stage are done.
      wait_asynccnt<3>();

      const float4 a = tile[phase][0][tid];
      const float4 b = tile[phase][1][tid];
      const float4 c = tile[phase][2][tid];
      // packed xyz triples: (a.x,a.y,a.z)(a.w,b.x,b.y)(b.z,b.w,c.x)(c.y,c.z,c.w)
      const float s0 = LS2 * (a.x * a.x + a.y * a.y + a.z * a.z);
      const float s1 = LS2 * (a.w * a.w + b.x * b.x + b.y * b.y);
      const float s2 = LS2 * (b.z * b.z + b.w * b.w + c.x * c.x);
      const float s3 = LS2 * (c.y * c.y + c.z * c.z + c.w * c.w);

      float4 r;
      r.x = tau_of_s(s0);
      r.y = tau_of_s(s1);
      r.z = tau_of_s(s2);
      r.w = tau_of_s(s3);
      ov[p] = r;

      phase ^= 1;
    }
    // Final stage's dangling prefetches are retired by S_ENDPGM's wait-idle.
  }

  // generic tail (n not divisible by 4; empty for 256^3)
  const long long base = npack * 4;
  for (long long i = base + g; i < n; i += stride) {
    const float x = k[i * 3 + 0], y = k[i * 3 + 1], z = k[i * 3 + 2];
    out[i] = tau_of_s(LS2 * (x * x + y * y + z * z));
  }
}

extern "C" void kernel_launch(void* const* d_in, const int* in_sizes, int n_in,
                              void* d_out, int out_size, void* d_ws, size_t ws_size,
                              hipStream_t stream) {
  (void)n_in; (void)out_size; (void)d_ws; (void)ws_size;
  const long long n     = (long long)in_sizes[0] / 3;  // number of k-vectors
  const long long npack = n / 4;

  long long want = (npack + TPB - 1) / TPB;
  if (want < 1) want = 1;
  // Grid-stride with ~4 packs/thread so the async pipeline actually runs ahead.
  int blocks = (int)(want > 4096 ? 4096 : want);

  mann_tau_kernel<<<blocks, TPB, 0, stream>>>(
      (const float*)d_in[0], (float*)d_out, npack, n);
}